// VariationalQuantumClassifier_35966056136959
// MI455X (gfx1250) — compile-verified
//
#include <hip/hip_runtime.h>
#include <math.h>

// ---------------------------------------------------------------------------
// VQC: angles = tanh(x @ W_in^T + b_in) * pi       [2048 x 12]   (WMMA f32)
//      12-qubit statevector sim (RY enc, 4x(RY,RZ,CNOT-ring))    (LDS)
//      out = zfeat @ W_out^T + b_out               [2048 x 4]    (fused)
// ---------------------------------------------------------------------------

typedef __attribute__((ext_vector_type(2))) float v2f;
typedef __attribute__((ext_vector_type(8))) float v8f;

#define NQ        12
#define DIM       4096          // 2^NQ
#define INPUT_DIM 64
#define NCLASS    4
#define NLAYERS   4

// ---------------------------------------------------------------------------
// Kernel 1: input projection with V_WMMA_F32_16X16X4_F32.
// One wave (32 threads) per 16-row tile of x. N dim = 16 (12 valid qubits,
// cols 12..15 masked to zero in the B fragment). K = 64 as 16 steps of 4.
//
// ISA fragment layouts (CDNA5 ISA):
//   A 16x4 f32 : lanes 0-15 -> M=0..15 with K={0,1} in v[0],v[1];
//                lanes 16-31 -> same M with K={2,3}.
//   B 4x16 f32 : lanes 0-15 -> N=0..15 with K={0,1}; lanes 16-31 -> K={2,3}.
//   C/D 16x16  : VGPR r, lanes 0-15 -> M=r, N=lane; lanes 16-31 -> M=r+8.
// ---------------------------------------------------------------------------
__global__ __launch_bounds__(32)
void vqc_angles_wmma(const float* __restrict__ x,
                     const float* __restrict__ W_in,
                     const float* __restrict__ b_in,
                     float* __restrict__ angles, int B)
{
    const int lane = threadIdx.x;          // 0..31
    const int tile = blockIdx.x;           // 16 rows of x per tile
    const int half = lane >> 4;            // 0: K={0,1}, 1: K={2,3}
    const int mn   = lane & 15;            // M for A, N for B

    int rowA = tile * 16 + mn;
    if (rowA > B - 1) rowA = B - 1;        // clamp (stores guarded below)
    const int   nclamp = (mn < NQ) ? mn : (NQ - 1);
    const float nmask  = (mn < NQ) ? 1.0f : 0.0f;

    v8f acc = {};
    #pragma unroll
    for (int kk = 0; kk < INPUT_DIM / 4; ++kk) {
        const int k0 = 4 * kk + 2 * half;
        v2f a, bm;
        a.x  = x[rowA * INPUT_DIM + k0];
        a.y  = x[rowA * INPUT_DIM + k0 + 1];
        // B = W_in^T : B[k][n] = W_in[n][k]; columns n>=12 zeroed.
        bm.x = W_in[nclamp * INPUT_DIM + k0]     * nmask;
        bm.y = W_in[nclamp * INPUT_DIM + k0 + 1] * nmask;
        // (neg_a, A, neg_b, B, c_mod, C, reuse_a, reuse_b)
        acc = __builtin_amdgcn_wmma_f32_16x16x4_f32(
                  false, a, false, bm, (short)0, acc, false, false);
    }

    const float PI_F = 3.14159274101257324f;  // float32(np.pi)
    if (mn < NQ) {
        #pragma unroll
        for (int r = 0; r < 8; ++r) {
            const int row = tile * 16 + r + 8 * half;
            if (row < B) {
                const float v = acc[r] + b_in[mn];
                angles[row * NQ + mn] = tanhf(v) * PI_F;
            }
        }
    }
}

// ---------------------------------------------------------------------------
// Kernel 2: one workgroup (256 threads = 8 waves) per sample.
// Full 32 KB complex statevector lives in LDS for the whole circuit.
// RZ (diagonal) is fused into the RY pair update; CNOTs are LDS swaps;
// PauliZ expectations + output projection are fused at the end.
// ---------------------------------------------------------------------------
__global__ __launch_bounds__(256)
void vqc_sim(const float* __restrict__ angles,
             const float* __restrict__ qw,
             const float* __restrict__ W_out,
             const float* __restrict__ b_out,
             float* __restrict__ out)
{
    __shared__ float2 st[DIM];     // 32 KB statevector
    __shared__ float  zred[NQ];    // cross-wave reduction of <Z_q>

    const int tid  = threadIdx.x;
    const int samp = blockIdx.x;

    if (tid < NQ) zred[tid] = 0.0f;
    for (int i = tid; i < DIM; i += 256)
        st[i] = make_float2(i == 0 ? 1.0f : 0.0f, 0.0f);
    __syncthreads();

    // ---- encoding: RY(angle[samp][q]) on each qubit (real 2x2) ----
    for (int q = 0; q < NQ; ++q) {
        const float th = angles[samp * NQ + q];
        const float cy = cosf(0.5f * th), sy = sinf(0.5f * th);
        const int mlo = (1 << q) - 1;
        for (int p = tid; p < DIM / 2; p += 256) {
            const int i0 = ((p & ~mlo) << 1) | (p & mlo);   // bit q = 0
            const int i1 = i0 | (1 << q);                   // bit q = 1
            const float2 a0 = st[i0], a1 = st[i1];
            st[i0] = make_float2(cy * a0.x - sy * a1.x, cy * a0.y - sy * a1.y);
            st[i1] = make_float2(sy * a0.x + cy * a1.x, sy * a0.y + cy * a1.y);
        }
        __syncthreads();
    }

    // ---- variational layers ----
    for (int l = 0; l < NLAYERS; ++l) {
        // fused RY(qw[l][q][0]) then RZ(qw[l][q][1]) per qubit
        for (int q = 0; q < NQ; ++q) {
            const float ty = qw[(l * NQ + q) * 2 + 0];
            const float tz = qw[(l * NQ + q) * 2 + 1];
            const float cy = cosf(0.5f * ty), sy = sinf(0.5f * ty);
            const float cz = cosf(0.5f * tz), sz = sinf(0.5f * tz);
            const int mlo = (1 << q) - 1;
            for (int p = tid; p < DIM / 2; p += 256) {
                const int i0 = ((p & ~mlo) << 1) | (p & mlo);
                const int i1 = i0 | (1 << q);
                const float2 a0 = st[i0], a1 = st[i1];
                // RY
                const float n0x = cy * a0.x - sy * a1.x;
                const float n0y = cy * a0.y - sy * a1.y;
                const float n1x = sy * a0.x + cy * a1.x;
                const float n1y = sy * a0.y + cy * a1.y;
                // RZ: amp(bit=0) *= (cz - i sz); amp(bit=1) *= (cz + i sz)
                st[i0] = make_float2(fmaf(n0y,  sz, n0x * cz),
                                     fmaf(-n0x, sz, n0y * cz));
                st[i1] = make_float2(fmaf(-n1y, sz, n1x * cz),
                                     fmaf(n1x,  sz, n1y * cz));
            }
            __syncthreads();
        }
        // CNOT ring: (0,1),(1,2),...,(10,11),(11,0) as LDS swaps
        for (int e = 0; e < NQ; ++e) {
            const int cq = e, tq = (e + 1) % NQ;
            const int lo = cq < tq ? cq : tq;
            const int hi = cq < tq ? tq : cq;
            const int mlo = (1 << lo) - 1;
            const int mhi = (1 << hi) - 1;
            for (int j = tid; j < DIM / 4; j += 256) {
                int idx = ((j   & ~mlo) << 1) | (j   & mlo);  // insert 0 @ lo
                idx     = ((idx & ~mhi) << 1) | (idx & mhi);  // insert 0 @ hi
                const int i0 = idx | (1 << cq);               // control=1, target=0
                const int i1 = i0  | (1 << tq);               // control=1, target=1
                const float2 t = st[i0];
                st[i0] = st[i1];
                st[i1] = t;
            }
            __syncthreads();
        }
    }

    // ---- <Z_q> = P(bit_q=0) - P(bit_q=1) ----
    float z[NQ];
    #pragma unroll
    for (int q = 0; q < NQ; ++q) z[q] = 0.0f;
    for (int i = tid; i < DIM; i += 256) {
        const float2 a = st[i];
        const float pr = a.x * a.x + a.y * a.y;
        #pragma unroll
        for (int q = 0; q < NQ; ++q)
            z[q] += ((i >> q) & 1) ? -pr : pr;
    }
    // wave32 reduce, then one LDS atomic per wave per qubit
    #pragma unroll
    for (int q = 0; q < NQ; ++q) {
        #pragma unroll
        for (int off = 16; off > 0; off >>= 1)
            z[q] += __shfl_down(z[q], off, 32);
    }
    if ((tid & 31) == 0) {
        #pragma unroll
        for (int q = 0; q < NQ; ++q) atomicAdd(&zred[q], z[q]);
    }
    __syncthreads();

    // ---- fused output projection: out = z @ W_out^T + b_out ----
    if (tid < NCLASS) {
        float acc = b_out[tid];
        #pragma unroll
        for (int q = 0; q < NQ; ++q) acc += W_out[tid * NQ + q] * zred[q];
        out[samp * NCLASS + tid] = acc;
    }
}

// ---------------------------------------------------------------------------
extern "C" void kernel_launch(void* const* d_in, const int* in_sizes, int n_in,
                              void* d_out, int out_size, void* d_ws, size_t ws_size,
                              hipStream_t stream)
{
    const float* x     = (const float*)d_in[0];  // [B, 64]
    const float* W_in  = (const float*)d_in[1];  // [12, 64]
    const float* b_in  = (const float*)d_in[2];  // [12]
    const float* qw    = (const float*)d_in[3];  // [4, 12, 2]
    const float* W_out = (const float*)d_in[4];  // [4, 12]
    const float* b_out = (const float*)d_in[5];  // [4]
    float*       out   = (float*)d_out;          // [B, 4]
    float*       angles = (float*)d_ws;          // [B, 12] scratch

    const int B = in_sizes[0] / INPUT_DIM;       // 2048
    const int tiles = (B + 15) / 16;

    vqc_angles_wmma<<<tiles, 32, 0, stream>>>(x, W_in, b_in, angles, B);
    vqc_sim<<<B, 256, 0, stream>>>(angles, qw, W_out, b_out, out);
}